// _ECELoss_6459630813868
// MI455X (gfx1250) — compile-verified
//
#include <hip/hip_runtime.h>
#include <hip/hip_bf16.h>

// ---------------------------------------------------------------------------
// ECE loss for logits[N,10], labels[N]  (N = 4M in harness)
// Memory-bound: ~176 MB -> ~7.6us floor @ 23.3 TB/s.
// Kernel 1: async global->LDS double-buffered streaming, per-lane-pair LDS
//           histograms (ds_add_f32, conflict degree <= 2), block partials.
// Kernel 2: partial reduction via V_WMMA_F32_16X16X4_F32 (ones-matrix trick),
//           deterministic fixed-order merge, scalar ECE finish.
// ---------------------------------------------------------------------------

typedef __attribute__((ext_vector_type(2))) float v2f;
typedef __attribute__((ext_vector_type(8))) float v8f;
typedef __attribute__((ext_vector_type(2))) int   v2i;

typedef __attribute__((address_space(1))) v2i* gptr_v2i;   // global b64 chunk
typedef __attribute__((address_space(3))) v2i* lptr_v2i;   // LDS b64 chunk

#define NBINS 15
#define NENT  48            // [0..14]=count  [16..30]=conf_sum  [32..46]=acc_sum
#define HSTRIDE 49          // 49 coprime with 64 banks -> conflict-free bases
#define TPB   256           // 8 waves (wave32)
#define ROWS_PER_TILE 256   // one row per thread per tile
#define CCLS  10

// ---- CDNA5 async global->LDS copy (ASYNCcnt path) -------------------------
__device__ __forceinline__ void async_copy_b64(const float* gsrc, float* lds_dst) {
#if __has_builtin(__builtin_amdgcn_global_load_async_to_lds_b64)
  __builtin_amdgcn_global_load_async_to_lds_b64(
      (gptr_v2i)gsrc, (lptr_v2i)lds_dst, 0, 0);
#else
  unsigned lds_off =
      (unsigned)(unsigned long long)(__attribute__((address_space(3))) void*)lds_dst;
  asm volatile("global_load_async_to_lds_b64 %0, %1, off"
               :: "v"(lds_off), "v"(gsrc) : "memory");
#endif
}

#if __has_builtin(__builtin_amdgcn_s_wait_asynccnt)
#define WAIT_ASYNC(n) __builtin_amdgcn_s_wait_asynccnt(n)
#else
#define WAIT_ASYNC(n) asm volatile("s_wait_asynccnt %0" :: "i"(n) : "memory")
#endif

// ---------------------------------------------------------------------------
// Kernel 1: streaming partial histograms
// ---------------------------------------------------------------------------
__global__ void __launch_bounds__(TPB)
ece_partial(const float* __restrict__ logits, const int* __restrict__ labels,
            float* __restrict__ partials, int nRows, int nTiles)
{
  __shared__ __align__(16) float tile[2][ROWS_PER_TILE * CCLS]; // 2 x 10240 B
  __shared__ float hist[(TPB / 2) * HSTRIDE];                   // 25088 B

  const int tid = threadIdx.x;
  for (int i = tid; i < (TPB / 2) * HSTRIDE; i += TPB) hist[i] = 0.0f;
  __syncthreads();

  float* h = &hist[(tid >> 1) * HSTRIDE];

  auto prefetch = [&](int buf, int tIdx) {
    const float* src = logits + (long long)tIdx * (ROWS_PER_TILE * CCLS);
#pragma unroll
    for (int k = 0; k < 5; ++k) {
      int j = k * TPB + tid;                  // b64 (8-byte) chunk index
      async_copy_b64(src + j * 2, &tile[buf][j * 2]);
    }
  };

  auto process = [&](const float* row, int rowIdx) {
    float x[CCLS];
    const float2* rp = (const float2*)row;    // 8B-aligned LDS reads
#pragma unroll
    for (int k = 0; k < 5; ++k) { float2 p = rp[k]; x[2*k] = p.x; x[2*k+1] = p.y; }
    float m = x[0]; int am = 0;
#pragma unroll
    for (int j = 1; j < CCLS; ++j) { if (x[j] > m) { m = x[j]; am = j; } }
    float s = 0.0f;
#pragma unroll
    for (int j = 0; j < CCLS; ++j) s += __expf(x[j] - m);   // v_exp_f32
    float conf = 1.0f / s;                                   // max softmax prob
    int lab = labels[rowIdx];
    float a = (am == lab) ? 1.0f : 0.0f;
    int bin = (int)ceilf(conf * (float)NBINS) - 1;           // (lower, upper]
    bin = min(max(bin, 0), NBINS - 1);
    atomicAdd(&h[bin], 1.0f);            // ds_add_f32, conflict degree <= 2
    atomicAdd(&h[16 + bin], conf);
    atomicAdd(&h[32 + bin], a);
  };

  int t0 = blockIdx.x;
  int buf = 0;
  if (t0 < nTiles) prefetch(0, t0);
  for (int t = t0; t < nTiles; t += gridDim.x) {
    int tn = t + gridDim.x;
    bool hasNext = (tn < nTiles);
    if (hasNext) { prefetch(buf ^ 1, tn); WAIT_ASYNC(5); }
    else         { WAIT_ASYNC(0); }
    __syncthreads();                       // all waves' async portions landed
    process(&tile[buf][tid * CCLS], t * ROWS_PER_TILE + tid);
    __syncthreads();                       // done reading before buf is reused
    buf ^= 1;
  }

  // tail rows (N % 256) handled by block 0 with guarded direct loads
  if (blockIdx.x == 0) {
    int base = nTiles * ROWS_PER_TILE;
    for (int r = base + tid; r < nRows; r += TPB) {
      float x[CCLS];
      const float* g = logits + (long long)r * CCLS;
#pragma unroll
      for (int j = 0; j < CCLS; ++j) x[j] = g[j];
      float m = x[0]; int am = 0;
#pragma unroll
      for (int j = 1; j < CCLS; ++j) { if (x[j] > m) { m = x[j]; am = j; } }
      float s = 0.0f;
#pragma unroll
      for (int j = 0; j < CCLS; ++j) s += __expf(x[j] - m);
      float conf = 1.0f / s;
      float a = (am == labels[r]) ? 1.0f : 0.0f;
      int bin = (int)ceilf(conf * (float)NBINS) - 1;
      bin = min(max(bin, 0), NBINS - 1);
      atomicAdd(&h[bin], 1.0f);
      atomicAdd(&h[16 + bin], conf);
      atomicAdd(&h[32 + bin], a);
    }
  }
  __syncthreads();

  // fixed-order tree reduce: 128 pair-histograms -> 1 (deterministic)
  for (int s = TPB / 4; s >= 1; s >>= 1) {
    for (int idx = tid; idx < s * NENT; idx += TPB) {
      int g2 = idx / NENT, e = idx % NENT;
      hist[g2 * HSTRIDE + e] += hist[(g2 + s) * HSTRIDE + e];
    }
    __syncthreads();
  }
  if (tid < NENT) partials[blockIdx.x * NENT + tid] = hist[tid];
}

// ---------------------------------------------------------------------------
// Kernel 2: WMMA-based reduction of per-block partials + final ECE
// D = ones(16x4) x B(4x16) + C  ==> every D row = column sums (exact f32)
// ---------------------------------------------------------------------------
__global__ void __launch_bounds__(256)
ece_final(const float* __restrict__ partials, float* __restrict__ out,
          int nPart, float invN)
{
  __shared__ float wpart[8 * NENT];
  __shared__ float sums[NENT];
  const int tid = threadIdx.x;
  const int wave = tid >> 5, lane = tid & 31;
  const int n = lane & 15, kh = lane >> 4;

#if __has_builtin(__builtin_amdgcn_wmma_f32_16x16x4_f32)
  for (int cg = 0; cg < 3; ++cg) {                // 3 groups of 16 entries
    v8f acc = {};
    for (int chunk = wave; chunk < nPart / 4; chunk += 8) {
      int p0 = chunk * 4 + kh * 2;                // B frag: K rows {p0, p0+1}
      v2f b; b.x = partials[(p0 + 0) * NENT + cg * 16 + n];
             b.y = partials[(p0 + 1) * NENT + cg * 16 + n];
      v2f a; a.x = 1.0f; a.y = 1.0f;              // ones matrix
      acc = __builtin_amdgcn_wmma_f32_16x16x4_f32(
          false, a, false, b, (short)0, acc, false, false);
    }
    if (lane < 16) wpart[wave * NENT + cg * 16 + lane] = acc[0]; // D row M=0
  }
#else
  for (int e = tid; e < NENT; e += 256) {
    float s = 0.0f;
    int w = tid >> 5;
    for (int p = w; p < nPart; p += 8) s += partials[p * NENT + e];
    wpart[0 * NENT + e] = s;                      // fallback: single slot
    for (int w2 = 1; w2 < 8; ++w2) wpart[w2 * NENT + e] = 0.0f;
  }
#endif
  __syncthreads();

  if (tid < NENT) {                               // fixed-order merge (det.)
    float s = 0.0f;
    for (int w = 0; w < 8; ++w) s += wpart[w * NENT + tid];
    sums[tid] = s;
  }
  __syncthreads();

  if (tid == 0) {
    float ece = 0.0f;
    for (int b = 0; b < NBINS; ++b) {
      float cnt = sums[b], cs = sums[16 + b], as_ = sums[32 + b];
      float safe = fmaxf(cnt, 1.0f);
      float gap = fabsf(cs / safe - as_ / safe);
      if (cnt > 0.0f) ece += gap * cnt * invN;
    }
    out[0] = ece;
  }
}

// ---------------------------------------------------------------------------
extern "C" void kernel_launch(void* const* d_in, const int* in_sizes, int n_in,
                              void* d_out, int out_size, void* d_ws, size_t ws_size,
                              hipStream_t stream) {
  const float* logits = (const float*)d_in[0];
  const int*   labels = (const int*)d_in[1];
  int N = in_sizes[1];                       // labels element count
  int nTiles = N / ROWS_PER_TILE;

  int nBlocks = 1024;                        // 8192 waves of streaming work
  int maxBlocksWs = (int)(ws_size / (NENT * sizeof(float)));
  if (nBlocks > maxBlocksWs) nBlocks = maxBlocksWs & ~7;  // keep /4 divisible
  if (nBlocks < 8) nBlocks = 8;

  float invN = 1.0f / (float)N;

  ece_partial<<<nBlocks, TPB, 0, stream>>>(logits, labels, (float*)d_ws, N, nTiles);
  ece_final<<<1, 256, 0, stream>>>((const float*)d_ws, (float*)d_out, nBlocks, invN);
}